// SVPFParamEstimator_61452392071939
// MI455X (gfx1250) — compile-verified
//
#include <hip/hip_runtime.h>
#include <hip/hip_bf16.h>
#include <math.h>

typedef __attribute__((ext_vector_type(2))) float v2f;
typedef __attribute__((ext_vector_type(4))) float v4f;
typedef __attribute__((ext_vector_type(8))) float v8f;

#define ST_CONST  (-0.95010864f)   // lgamma(4.5)-lgamma(4)-0.5*log(8*pi)
#define LOG_N     (4.1588831f)     // log(64)
#define NU        (8.0f)

__device__ __forceinline__ float rcpf(float x) { return __builtin_amdgcn_rcpf(x); }
__device__ __forceinline__ float rsqf(float x) { return __builtin_amdgcn_rsqf(x); }

__device__ __forceinline__ float wave_reduce_sum(float v) {
#pragma unroll
  for (int m = 16; m >= 1; m >>= 1) v += __shfl_xor(v, m, 32);
  return v;
}
__device__ __forceinline__ float wave_reduce_max(float v) {
#pragma unroll
  for (int m = 16; m >= 1; m >>= 1) v = fmaxf(v, __shfl_xor(v, m, 32));
  return v;
}

// clip(grad_prior + grad_lik, -10, 10) for one particle
__device__ __forceinline__ float grad_logp(float hh, float tm, float y2,
                                           float inv_sz2) {
  float vol = __expf(0.5f * hh);
  float A   = y2 * rcpf(vol * vol * NU + 1e-8f);
  float gl  = fmaf(4.5f * A, rcpf(1.0f + A), -0.5f);
  float gp  = -(hh - tm) * inv_sz2;
  return fminf(fmaxf(gp + gl, -10.0f), 10.0f);
}

// Each wave owns a 16x64 block of M2[i,j] = K(i,j)*g[j] + gradK(i,j).
// rowsum(M2) = M2 x ones computed with V_WMMA_F32_16X16X4_F32 (full f32).
// A layout (16x4 f32): lanes 0-15 -> K=0,1 ; lanes 16-31 -> K=2,3.
// s_hg interleaves h/g so one ds_load_b128 fetches a chunk's 4 operands.
__device__ __forceinline__ void stein_rowsum(const float* s_hg,
                                             float* s_phi, float inv_bwsq,
                                             int lane, int wv) {
  const int   row  = (wv << 4) + (lane & 15);
  const int   half = lane >> 4;                  // 0 or 1 (j0 = 4c + 2*half)
  const float hi   = s_hg[2 * row];
  const v4f*  hg4  = (const v4f*)s_hg;           // hg4[q]={h[2q],g[2q],h[2q+1],g[2q+1]}
  v8f acc = {};
  const v2f bones = {1.0f, 1.0f};                // B = ones(4x16)

  // 1-deep pipeline: one b128 prefetch per chunk, a full body ahead of use.
  v4f cur = hg4[half];
#pragma unroll
  for (int c = 0; c < 16; ++c) {
    v4f nxt = cur;
    if (c < 15) nxt = hg4[2 * (c + 1) + half];
    // m = K*g + gradK with ds = d/bw_sq:
    //   base = 1 + d*ds ; K = rsqrt(base) (base >= 1) ; gradK = -ds*K^3
    float d0  = hi - cur.x;
    float d1  = hi - cur.z;
    float ds0 = d0 * inv_bwsq;
    float ds1 = d1 * inv_bwsq;
    float k0  = rsqf(fmaf(d0, ds0, 1.0f));
    float k1  = rsqf(fmaf(d1, ds1, 1.0f));
    float t0  = ds0 * (k0 * k0 * k0);
    float t1  = ds1 * (k1 * k1 * k1);
    float m0  = fmaf(cur.y, k0, -t0);
    float m1  = fmaf(cur.w, k1, -t1);
    v2f a = {m0, m1};
    acc = __builtin_amdgcn_wmma_f32_16x16x4_f32(
        false, a, false, bones, (short)0, acc, false, false);
    cur = nxt;
  }
  // C/D layout: VGPR r, lanes 0-15 -> M=r ; lanes 16-31 -> M=8+r.
  // B=ones => all columns equal the row sum; peel from lanes 0 and 16.
  if (lane == 0) {
#pragma unroll
    for (int r = 0; r < 8; ++r) s_phi[(wv << 4) + r] = acc[r];
  } else if (lane == 16) {
#pragma unroll
    for (int r = 0; r < 8; ++r) s_phi[(wv << 4) + 8 + r] = acc[r];
  }
}

__global__ __launch_bounds__(128)
void svpf_kernel(const float* __restrict__ y,
                 const float* __restrict__ rho_logit_p,
                 const float* __restrict__ log_sigma_p,
                 const float* __restrict__ mu_p,
                 const float* __restrict__ h_noise,
                 const float* __restrict__ eps_half,
                 int T,
                 const int* __restrict__ burn_in_p,
                 float* __restrict__ out) {
  __shared__ __align__(16) float s_hg[128];  // [2j]=h_j (state), [2j+1]=g_j
  __shared__ float s_tm[64];                 // transition mean (grad_prior)
  __shared__ float s_logp[64];
  __shared__ float s_phi[64];
  __shared__ float s_red[2];                 // [0]=log_lik_t, [1]=bw_sq

  const int tid  = (int)threadIdx.x;
  const int lane = tid & 31;
  const int wv   = tid >> 5;
  const int burn_in = burn_in_p[0];

  // ---- parameter transforms (uniform across threads, outside the scan) ----
  const float rho_logit = rho_logit_p[0];
  const float log_sigma = log_sigma_p[0];
  const float mu        = mu_p[0];
  const float rho     = 0.5f + 0.499f * (1.0f / (1.0f + __expf(-rho_logit)));
  const float sigma_z = __logf(1.0f + __expf(log_sigma)) + 0.001f;
  const float inv_sz2 = 1.0f / (sigma_z * sigma_z + 1e-8f);
  const float h_std   = sigma_z / sqrtf(1.0f - rho * rho + 1e-6f);

  float h_reg = 0.0f, tm_reg = 0.0f;  // per-particle register mirrors (tid<64)
  if (tid < 64) {
    h_reg = fmaf(h_std, h_noise[tid], mu);
    s_hg[2 * tid] = h_reg;
  }
  __syncthreads();

  float llsum = 0.0f;

  for (int t = 0; t < T; ++t) {
    const float y_t = y[t];
    const float y2  = y_t * y_t;

    // ---- stage A: propagate + per-particle Student-t log-lik ----
    if (tid < 64) {
      float e = (tid < 32) ? eps_half[t * 32 + tid]
                           : -eps_half[t * 32 + (tid - 32)];  // antithetic
      float tm = fmaf(rho, h_reg - mu, mu);
      float hp = fmaf(sigma_z, e, tm);
      tm_reg = tm;
      h_reg  = hp;                                  // hh := h_pred
      s_tm[tid]     = tm;
      s_hg[2 * tid] = hp;
      float vol = __expf(0.5f * hp) + 1e-8f;        // vol_pred + 1e-8
      float z   = y_t * rcpf(vol);
      s_logp[tid] = ST_CONST - __logf(vol)
                  - 4.5f * __logf(fmaf(z * z, 1.0f / NU, 1.0f));
    }
    __syncthreads();

    // ---- stage B: wave 0 reductions || waves 2-3 grad for Stein iter 0 ----
    // (wave 0 reads .x halves; waves 2-3 write .y halves: disjoint bytes)
    if (wv == 0) {
      float a = s_logp[lane], b = s_logp[lane + 32];
      float mx = wave_reduce_max(fmaxf(a, b));
      float se = wave_reduce_sum(__expf(a - mx) + __expf(b - mx));
      float h1 = s_hg[2 * lane], h2 = s_hg[2 * (lane + 32)];
      float mean = wave_reduce_sum(h1 + h2) * (1.0f / 64.0f);
      float d1 = h1 - mean, d2 = h2 - mean;
      float var = wave_reduce_sum(d1 * d1 + d2 * d2) * (1.0f / 64.0f);
      if (lane == 0) {
        s_red[0] = __logf(se) + mx - LOG_N;
        float bw = sqrtf(var) + 0.1f;               // std is stop_gradient'd
        s_red[1] = fmaf(bw, bw, 1e-8f);
      }
    } else if (tid >= 64) {
      int p = tid - 64;
      s_hg[2 * p + 1] = grad_logp(s_hg[2 * p], s_tm[p], y2, inv_sz2);
    }
    __syncthreads();
    const float inv_bwsq = rcpf(s_red[1]);

    // ---- Stein iteration 0: WMMA row sums -> s_phi ----
    stein_rowsum(s_hg, s_phi, inv_bwsq, lane, wv);
    __syncthreads();

    // ---- update + grad for Stein iteration 1 (single b64 store) ----
    if (tid < 64) {
      float hh = fmaf(s_phi[tid], 0.1f / 64.0f, h_reg);
      h_reg = hh;
      v2f hg = {hh, grad_logp(hh, tm_reg, y2, inv_sz2)};
      *(v2f*)(s_hg + 2 * tid) = hg;
    }
    __syncthreads();

    // ---- Stein iteration 1: WMMA row sums -> s_phi ----
    stein_rowsum(s_hg, s_phi, inv_bwsq, lane, wv);
    __syncthreads();

    // ---- final update + per-particle vol (single b64 store) ----
    if (tid < 64) {
      float hh = fmaf(s_phi[tid], 0.1f / 64.0f, h_reg);
      h_reg = hh;
      v2f hv = {hh, __expf(0.5f * hh)};   // .x = state for next step
      *(v2f*)(s_hg + 2 * tid) = hv;
    }
    __syncthreads();

    // ---- vol_est reduction + log-lik accumulation ----
    if (wv == 0) {
      float sum = wave_reduce_sum(s_hg[2 * lane + 1] + s_hg[2 * (lane + 32) + 1]);
      if (lane == 0) out[1 + t] = sum * (1.0f / 64.0f);
    }
    if (tid == 0 && t >= burn_in) llsum += s_red[0];
    // no trailing barrier: next stage A writes only .x / s_tm / s_logp and a
    // barrier follows before anyone re-reads .y halves or re-writes s_red.
  }

  if (tid == 0) out[0] = -llsum / (float)(T - burn_in);
}

extern "C" void kernel_launch(void* const* d_in, const int* in_sizes, int n_in,
                              void* d_out, int out_size, void* d_ws, size_t ws_size,
                              hipStream_t stream) {
  (void)n_in; (void)out_size; (void)d_ws; (void)ws_size;
  const float* y         = (const float*)d_in[0];
  const float* rho_logit = (const float*)d_in[1];
  const float* log_sigma = (const float*)d_in[2];
  const float* mu_param  = (const float*)d_in[3];
  const float* h_noise   = (const float*)d_in[4];
  const float* eps_half  = (const float*)d_in[5];
  // d_in[6] = truncate_every: stop_gradient is identity in the forward pass.
  const int*   burn_in_p = (const int*)d_in[7];
  const int T = in_sizes[0];

  svpf_kernel<<<1, 128, 0, stream>>>(y, rho_logit, log_sigma, mu_param,
                                     h_noise, eps_half, T, burn_in_p,
                                     (float*)d_out);
}